// SparseLayer_46806553591821
// MI455X (gfx1250) — compile-verified
//
#include <hip/hip_runtime.h>

#define IN_DIM  20000
#define OUT_DIM 5000
#define NNZ     320000
#define BATCH   64

typedef __attribute__((ext_vector_type(2))) float v2f;
typedef __attribute__((ext_vector_type(8))) float v8f;

// ---------------------------------------------------------------------------
// Kernel 1: zero the transposed accumulator outT[OUT_DIM][BATCH].
// Grid sized exactly: OUT_DIM*BATCH = 320000 = 1250 * 256.
// ---------------------------------------------------------------------------
__global__ void k_zero_outT(float* __restrict__ outT) {
  int i = blockIdx.x * blockDim.x + threadIdx.x;
  outT[i] = 0.0f;
}

// ---------------------------------------------------------------------------
// Kernel 2: transpose x[BATCH][IN_DIM] -> xT[IN_DIM][BATCH] so each sparse
// gather of x[:, r] becomes one contiguous 256B run. Writes coalesced;
// reads are strided but x (5.1 MB) lives in L2. Exact grid: 1,280,000/256.
// ---------------------------------------------------------------------------
__global__ void k_transpose_x(const float* __restrict__ x, float* __restrict__ xT) {
  int i = blockIdx.x * blockDim.x + threadIdx.x;   // over IN_DIM*BATCH
  int r = i >> 6;
  int b = i & 63;
  xT[i] = x[(size_t)b * IN_DIM + r];
}

// ---------------------------------------------------------------------------
// Kernel 3: sparse scatter. One wave32 handles 8 nonzeros; lanes 0..7 load
// the COO stream coalesced and broadcast via __shfl. Each lane covers 2
// batch elements (float2). Accumulation via HW f32 global atomics into the
// transposed outT (contiguous 256B per nonzero -> 4 cachelines, not 64).
// Exact grid: NNZ / (8 waves * 8 nnz) = 5000 blocks of 256.
// ---------------------------------------------------------------------------
__global__ void __launch_bounds__(256) k_scatter(const int2*  __restrict__ idx,
                                                 const float* __restrict__ wgt,
                                                 const float* __restrict__ xT,
                                                 float*       __restrict__ outT) {
  const int lane = threadIdx.x & 31;
  const int gw   = blockIdx.x * (blockDim.x >> 5) + (threadIdx.x >> 5);
  const int base = gw * 8;

  // Stream prefetch of upcoming index/weight data (gfx1250 global_prefetch_b8).
  if (base + 2048 < NNZ) {
    __builtin_prefetch(&idx[base + 2048], 0, 0);
    __builtin_prefetch(&wgt[base + 2048], 0, 0);
  }

  // Lanes 0..7 carry the 8 nonzeros (replicated x4 across the wave; coalesced).
  int2  rc  = idx[base + (lane & 7)];
  float wv8 = wgt[base + (lane & 7)];

#pragma unroll
  for (int k = 0; k < 8; ++k) {
    int   r  = __shfl(rc.x, k);
    int   c  = __shfl(rc.y, k);
    float wv = __shfl(wv8, k);

    v2f xv = *(const v2f*)(xT + (size_t)r * 64 + 2 * lane);
    float* dst = outT + (size_t)c * 64 + 2 * lane;
    unsafeAtomicAdd(dst,     wv * xv.x);   // global_atomic_add_f32
    unsafeAtomicAdd(dst + 1, wv * xv.y);
  }
}

// ---------------------------------------------------------------------------
// Kernel 4: epilogue out[b][c] = outT[c][b] + bias[c] via WMMA transpose.
// D = sum_j S_j(16x4) x B_j(4x16) + C, S_j[m][k] = (m == 4j+k), C = bias
// broadcast. D lands in C/D layout (rows across lanes) -> coalesced stores.
// One wave per 16x16 tile; 4 x V_WMMA_F32_16X16X4_F32 per tile.
// A layout (32-bit 16x4): lane M = lane&15; VGPR0/1 = K {0,1} (lanes 0-15)
// or K {2,3} (lanes 16-31). B mirrors A with N across lanes.
// ---------------------------------------------------------------------------
__global__ void __launch_bounds__(256) k_finalize(const float* __restrict__ outT,
                                                  const float* __restrict__ bias,
                                                  float*       __restrict__ out) {
  const int lane = threadIdx.x & 31;
  const int tile = blockIdx.x * (blockDim.x >> 5) + (threadIdx.x >> 5);
  const int CT   = (OUT_DIM + 15) / 16;          // 313 column tiles
  if (tile >= CT * 4) return;                    // wave-uniform exit

  const int ct    = tile >> 2;
  const int bbase = (tile & 3) * 16;             // batch tile base (0,16,32,48)
  const int n     = lane & 15;                   // N index (output column)
  const int half  = lane >> 4;                   // lane-half selects K pair
  const int kbase = half * 2;
  const int c     = ct * 16 + n;
  const int cc    = c < OUT_DIM ? c : OUT_DIM - 1;  // clamp loads, EXEC stays full

  float bv = bias[cc];
  v8f acc;
#pragma unroll
  for (int p = 0; p < 8; ++p) acc[p] = bv;       // C = bias broadcast

  const float* src = outT + (size_t)cc * 64 + bbase + kbase;
#pragma unroll
  for (int j = 0; j < 4; ++j) {
    v2f B = *(const v2f*)(src + 4 * j);          // B_j[k][n] = outT[c][bbase+4j+k]
    v2f A;
    A.x = (n == 4 * j + kbase)     ? 1.0f : 0.0f;  // S_j[m][kbase]
    A.y = (n == 4 * j + kbase + 1) ? 1.0f : 0.0f;  // S_j[m][kbase+1]
    acc = __builtin_amdgcn_wmma_f32_16x16x4_f32(
        /*neg_a=*/false, A, /*neg_b=*/false, B,
        /*c_mod=*/(short)0, acc, /*reuse_a=*/false, /*reuse_b=*/false);
  }

  if (c < OUT_DIM) {
#pragma unroll
    for (int p = 0; p < 8; ++p) {
      int m = bbase + p + half * 8;              // batch row from C/D layout
      out[(size_t)m * OUT_DIM + c] = acc[p];     // coalesced 64B row stores
    }
  }
}

// ---------------------------------------------------------------------------
// Host launcher. Inputs: x, nonzero_indices[NNZ][2], weights, bias.
// Workspace: xT (5.12 MB) then outT (1.28 MB).
// ---------------------------------------------------------------------------
extern "C" void kernel_launch(void* const* d_in, const int* in_sizes, int n_in,
                              void* d_out, int out_size, void* d_ws, size_t ws_size,
                              hipStream_t stream) {
  const float* x    = (const float*)d_in[0];
  const int2*  idx  = (const int2*)d_in[1];
  const float* wgt  = (const float*)d_in[2];
  const float* bias = (const float*)d_in[3];
  float*       out  = (float*)d_out;

  float* xT   = (float*)d_ws;
  float* outT = (float*)((char*)d_ws + (size_t)IN_DIM * BATCH * sizeof(float));

  k_zero_outT<<<(OUT_DIM * BATCH) / 256, 256, 0, stream>>>(outT);
  k_transpose_x<<<(IN_DIM * BATCH) / 256, 256, 0, stream>>>(x, xT);
  k_scatter<<<NNZ / 64, 256, 0, stream>>>(idx, wgt, xT, outT);

  int tiles = ((OUT_DIM + 15) / 16) * 4;         // 1252 wave-tiles
  k_finalize<<<(tiles + 7) / 8, 256, 0, stream>>>(outT, bias, out);
}